// DeepDynamicalSystemLayer_42125039239869
// MI455X (gfx1250) — compile-verified
//
#include <hip/hip_runtime.h>
#include <stdint.h>

// Kinematic bicycle rollout, B=131072 agents, K=64 Euler steps.
// Memory-bound (~195 MB traffic, ~8.4 us floor at 23.3 TB/s).
// CDNA5 path: TENSOR_LOAD_TO_LDS stages each block's control tile into LDS
// with hardware row padding (conflict-free wave32 ds_load_b64 afterwards).

#define KSTEPS      64
#define CTRL_DIM    2
#define AG_PER_BLK  64              // 2 waves of 32 (wave32)
#define ROW_DATA_DW (KSTEPS * CTRL_DIM)   // 128 dwords of controls per agent
#define ROW_DW      (ROW_DATA_DW + 2)     // +2 dword TDM pad -> bank-conflict-free

typedef __attribute__((ext_vector_type(4))) unsigned int v4u;
typedef __attribute__((ext_vector_type(8))) int          v8i;
typedef __attribute__((ext_vector_type(4))) int          v4i;

#if defined(__has_builtin)
#if __has_builtin(__builtin_amdgcn_tensor_load_to_lds) && \
    __has_builtin(__builtin_amdgcn_s_wait_tensorcnt)
#define USE_TDM 1
#endif
#endif
#ifndef USE_TDM
#define USE_TDM 0
#endif

__global__ __launch_bounds__(AG_PER_BLK)
void bicycle_rollout_kernel(const float* __restrict__ init_state,   // [B,4]
                            const float* __restrict__ controls,     // [B,K,2]
                            const float* __restrict__ dt_ptr,       // [1]
                            const float* __restrict__ pars,         // [B,2]
                            float* __restrict__ out)                // [B,K,4]
{
    __shared__ __align__(16) unsigned int lds_ctrl[AG_PER_BLK * ROW_DW]; // 33,280 B

    const int t     = threadIdx.x;
    const int b0    = blockIdx.x * AG_PER_BLK;
    const int agent = b0 + t;

#if USE_TDM
    // One TDM descriptor per block: 64 agents x 512B rows, global row stride
    // 512B, LDS row padded by 2 dwords (row stride 130 dwords => reads from
    // lane b hit bank (2b + 2k) % 64 -> conflict-free ds_load_b64).
    if (t < 32) {  // single wave issues the DMA (EXEC ignored by TDM)
        const uint64_t ga =
            (uint64_t)(uintptr_t)(controls + (uint64_t)b0 * ROW_DATA_DW);
        // low 32 bits of a flat shared pointer == LDS byte address
        const unsigned int lba = (unsigned int)(uintptr_t)(void*)&lds_ctrl[0];

        v4u g0;
        g0[0] = 1u;                                   // count=1 (valid user D#)
        g0[1] = lba;                                  // lds_addr
        g0[2] = (unsigned int)(ga & 0xFFFFFFFFu);     // global_addr[31:0]
        g0[3] = (unsigned int)((ga >> 32) & 0x1FFFFFFu) | (2u << 30); // addr[56:32] | type=2

        v8i g1;
        g1[0] = (2 << 16)      // data_size = 4 bytes
              | (1 << 20)      // pad_enable
              | (6 << 22)      // pad_interval: 128 dwords
              | (1 << 25);     // pad_amount:   2 dwords
        g1[1] = (ROW_DATA_DW & 0xFFFF) << 16;         // tensor_dim0 lo16 (128)
        g1[2] = (AG_PER_BLK << 16);                   // tensor_dim0 hi | tensor_dim1 lo (64)
        g1[3] = (ROW_DATA_DW << 16);                  // tensor_dim1 hi | tile_dim0 (128)
        g1[4] = AG_PER_BLK;                           // tile_dim1 (64) | tile_dim2=0
        g1[5] = ROW_DATA_DW;                          // tensor_dim0_stride lo32 (128 elems)
        g1[6] = 0;                                    // stride0 hi | stride1 lo
        g1[7] = 0;                                    // stride1 hi

        v4i g2 = {0, 0, 0, 0};                        // 2D tensor: groups 2/3 unused
        v4i g3 = {0, 0, 0, 0};
        v8i g4 = {0, 0, 0, 0, 0, 0, 0, 0};            // extra group (6-arg toolchain), unused

        __builtin_amdgcn_tensor_load_to_lds(g0, g1, g2, g3, g4, 0);
        __builtin_amdgcn_s_wait_tensorcnt(0);
    }
    asm volatile("" ::: "memory");   // LDS written behind the compiler's back
    __syncthreads();
#else
    // Fallback: coalesced manual stage into the same padded layout.
    for (int i = t; i < AG_PER_BLK * ROW_DATA_DW; i += AG_PER_BLK) {
        const int a = i >> 7;        // / 128
        const int j = i & 127;       // % 128  (coalesced fastest index)
        lds_ctrl[a * ROW_DW + j] =
            ((const unsigned int*)controls)[(uint64_t)(b0 + a) * ROW_DATA_DW + j];
    }
    __syncthreads();
#endif

    // Per-agent state: fully coalesced b128 load (lanes are consecutive agents).
    const float dt = dt_ptr[0];
    float4 s = ((const float4*)init_state)[agent];
    const float2 lw   = ((const float2*)pars)[agent];   // [length, width]
    const float  invL = 1.0f / lw.x;

    const float2* __restrict__ mc =
        (const float2*)&lds_ctrl[t * ROW_DW];           // conflict-free ds_load_b64
    float4* __restrict__ mo = ((float4*)out) + (uint64_t)agent * KSTEPS;

#pragma unroll 8
    for (int k = 0; k < KSTEPS; ++k) {
        const float2 u = mc[k];          // [accel, steer]
        // Euler derivatives from the *old* state (matches jax.lax.scan step).
        const float cy = __cosf(s.z);    // v_cos_f32 (yaw stays O(1) rad)
        const float sy = __sinf(s.z);    // v_sin_f32
        const float tn = __tanf(u.y);    // v_sin*rcp(v_cos), steer ~0.05 rad
        const float dx = s.w * cy;
        const float dy = s.w * sy;
        const float dz = s.w * tn * invL;
        const float dv = u.x;
        s.x += dt * dx;
        s.y += dt * dy;
        s.z += dt * dz;
        s.w += dt * dv;
        mo[k] = s;                       // global_store_b128; 8 steps fill a line
    }
}

extern "C" void kernel_launch(void* const* d_in, const int* in_sizes, int n_in,
                              void* d_out, int out_size, void* d_ws, size_t ws_size,
                              hipStream_t stream) {
    const float* init_state = (const float*)d_in[0];   // [B,4]
    const float* controls   = (const float*)d_in[1];   // [B,K,2]
    const float* dt_ptr     = (const float*)d_in[2];   // scalar
    const float* pars       = (const float*)d_in[3];   // [B,2]
    float*       out        = (float*)d_out;           // [B,K,4]

    const int n_agents = in_sizes[0] / 4;              // 131072
    const int blocks   = n_agents / AG_PER_BLK;        // 2048

    bicycle_rollout_kernel<<<blocks, AG_PER_BLK, 0, stream>>>(
        init_state, controls, dt_ptr, pars, out);
}